// MODEL_1013612282505
// MI455X (gfx1250) — compile-verified
//
#include <hip/hip_runtime.h>
#include <hip/hip_bf16.h>

typedef float v2f __attribute__((ext_vector_type(2)));
typedef float v8f __attribute__((ext_vector_type(8)));

#define TT      2048
#define KS      256
#define HSTRIDE 260   // floats; 260 % 64 == 4 -> conflict-free A-frag b64 loads

__global__ void __launch_bounds__(576, 1)
rnn_scan_wmma(const float* __restrict__ x,       // (128, 2048, 32)
              const float* __restrict__ state0,  // (128, 256)
              const float* __restrict__ W_xh,    // (256, 32)
              const float* __restrict__ W_hh,    // (256, 256)
              const float* __restrict__ b_hh,    // (256)
              const float* __restrict__ W_hy,    // (32, 256)
              const float* __restrict__ b_hy,    // (32)
              float* __restrict__ y_out,         // (128, 2048, 32)
              float* __restrict__ state_out)     // (128, 256)
{
    __shared__ float h_lds[16 * HSTRIDE];

    const int tid  = threadIdx.x;
    const int wave = tid >> 5;
    const int lane = tid & 31;
    const int ln   = lane & 15;   // N column (B/C/D) or M row (A) within tile
    const int hi   = lane >> 4;   // lane half -> K-pair select
    const int row0 = blockIdx.x * 16;

    // Per-lane base pointer for A fragments (M = ln, K-pair offset = 2*hi)
    const float* aptr = &h_lds[ln * HSTRIDE + 2 * hi];

    if (wave < 16) {
        // ----- h wave: owns h columns [16*wave, 16*wave+16) -----
        const int n = wave * 16 + ln;

        // B fragments of W_hh^T for this column tile: B[k][n] = W_hh[n][k]
        v2f bhh[64];
#pragma unroll
        for (int c = 0; c < 64; ++c)
            bhh[c] = *(const v2f*)(W_hh + n * KS + 4 * c + 2 * hi);
        v2f bxh[8];
#pragma unroll
        for (int c = 0; c < 8; ++c)
            bxh[c] = *(const v2f*)(W_xh + n * 32 + 4 * c + 2 * hi);
        const float bias = b_hh[n];

        // Seed LDS h with state0 (this wave's 16 columns, all 16 rows)
#pragma unroll
        for (int j = 0; j < 8; ++j) {
            int m = j + 8 * hi;
            h_lds[m * HSTRIDE + n] = state0[(size_t)(row0 + m) * KS + n];
        }

        const size_t xbase = ((size_t)(row0 + ln) * TT) * 32 + 2 * hi;
        float hn[8];

        __syncthreads();

#pragma unroll 1
        for (int t = 0; t < TT; ++t) {
            // Stream x ahead of the serial chain
            int tp = t + 8; if (tp >= TT) tp = TT - 1;
            __builtin_prefetch(x + xbase + (size_t)tp * 32, 0, 0);

            v2f xa[8];
#pragma unroll
            for (int c = 0; c < 8; ++c)
                xa[c] = *(const v2f*)(x + xbase + (size_t)t * 32 + 4 * c);

            v8f acc[4] = {};
            // h_{t-1} @ W_hh^T : 64 chained f32 WMMAs, 4 accumulators
#pragma unroll
            for (int c = 0; c < 64; ++c) {
                v2f a = *(const v2f*)(aptr + 4 * c);
                acc[c & 3] = __builtin_amdgcn_wmma_f32_16x16x4_f32(
                    false, a, false, bhh[c], (short)0, acc[c & 3], false, false);
            }
            // x_t @ W_xh^T : 8 WMMAs
#pragma unroll
            for (int c = 0; c < 8; ++c) {
                acc[c & 3] = __builtin_amdgcn_wmma_f32_16x16x4_f32(
                    false, xa[c], false, bxh[c], (short)0, acc[c & 3], false, false);
            }
            v8f pre = acc[0] + acc[1] + acc[2] + acc[3];

            // h = tanh(pre + b) via stable exp form (TRANS co-executes w/ WMMA)
#pragma unroll
            for (int j = 0; j < 8; ++j) {
                float p = pre[j] + bias;
                float e = __expf(-2.0f * fabsf(p));
                float r = (1.0f - e) / (1.0f + e);
                hn[j] = copysignf(r, p);
            }

            __syncthreads();   // everyone done reading h_{t-1}
#pragma unroll
            for (int j = 0; j < 8; ++j)
                h_lds[(j + 8 * hi) * HSTRIDE + n] = hn[j];
            __syncthreads();   // h_t published
        }

        // state_last = h_{T-1}
#pragma unroll
        for (int j = 0; j < 8; ++j)
            state_out[(size_t)(row0 + j + 8 * hi) * KS + n] = hn[j];
    } else {
        // ----- y wave: computes y_{t-1} while h waves compute h_t -----
        const int yt = wave - 16;      // y tile 0 or 1
        const int n  = yt * 16 + ln;   // output column

        v2f bhy[64];
#pragma unroll
        for (int c = 0; c < 64; ++c)
            bhy[c] = *(const v2f*)(W_hy + n * KS + 4 * c + 2 * hi);
        const float bias = b_hy[n];

        auto emit_y = [&](int tIdx) {
            v8f acc[4] = {};
#pragma unroll
            for (int c = 0; c < 64; ++c) {
                v2f a = *(const v2f*)(aptr + 4 * c);
                acc[c & 3] = __builtin_amdgcn_wmma_f32_16x16x4_f32(
                    false, a, false, bhy[c], (short)0, acc[c & 3], false, false);
            }
            v8f yv = acc[0] + acc[1] + acc[2] + acc[3];
#pragma unroll
            for (int j = 0; j < 8; ++j)
                y_out[((size_t)(row0 + j + 8 * hi) * TT + tIdx) * 32 + n] =
                    yv[j] + bias;
        };

        __syncthreads();
#pragma unroll 1
        for (int t = 0; t < TT; ++t) {
            if (t > 0) emit_y(t - 1);  // reads h_{t-1}, before overwrite
            __syncthreads();
            __syncthreads();
        }
        emit_y(TT - 1);
    }
}

extern "C" void kernel_launch(void* const* d_in, const int* in_sizes, int n_in,
                              void* d_out, int out_size, void* d_ws, size_t ws_size,
                              hipStream_t stream) {
    (void)in_sizes; (void)n_in; (void)out_size; (void)d_ws; (void)ws_size;
    const float* x     = (const float*)d_in[0];
    const float* s0    = (const float*)d_in[1];
    const float* W_xh  = (const float*)d_in[2];
    const float* W_hh  = (const float*)d_in[3];
    const float* b_hh  = (const float*)d_in[4];
    const float* W_hy  = (const float*)d_in[5];
    const float* b_hy  = (const float*)d_in[6];

    float* out       = (float*)d_out;
    float* y_out     = out;                                   // 16*8*2048*32
    float* state_out = out + (size_t)16 * 8 * 2048 * 32;      // 16*8*1*256

    rnn_scan_wmma<<<8, 576, 0, stream>>>(x, s0, W_xh, W_hh, b_hh, W_hy, b_hy,
                                         y_out, state_out);
}